// Net_317827580689
// MI455X (gfx1250) — compile-verified
//
#include <hip/hip_runtime.h>

#define D_IN  128
#define D_HID 16
#define D_OUT 32

typedef __attribute__((ext_vector_type(2))) float v2f;
typedef __attribute__((ext_vector_type(8))) float v8f;

// ---------------------------------------------------------------- zero
__global__ void gcn_zero(float* __restrict__ p, int n) {
    int i = blockIdx.x * blockDim.x + threadIdx.x;
    if (i < n) p[i] = 0.0f;
}

// ---------------------------------------------------------------- layer-1 GEMM
// H1[n x 16] = F[n x 128] @ W1[128 x 16], full fp32 via V_WMMA_F32_16X16X4_F32.
// One wave32 owns a 16-row tile; K reduced in 32 steps of 4.
__global__ __launch_bounds__(256) void gcn_gemm_l1(const float* __restrict__ F,
                                                   const float* __restrict__ W1,
                                                   float* __restrict__ H1,
                                                   int n_nodes) {
    int wave = (blockIdx.x * blockDim.x + threadIdx.x) >> 5;
    int lane = threadIdx.x & 31;
    int half = lane >> 4;      // 0: K={k,k+1}, 1: K={k+2,k+3}
    int l    = lane & 15;
    int row0 = wave * 16;
    if (row0 >= n_nodes) return;             // wave-uniform: EXEC stays all-1s

    int arow = row0 + l;
    if (arow >= n_nodes) arow = n_nodes - 1; // tail guard (N_NODES%16==0 anyway)
    const float* frow = F + (size_t)arow * D_IN;

    v8f c = {};
#pragma unroll
    for (int k = 0; k < D_IN; k += 4) {
        int kk = k + 2 * half;
        v2f a; a.x = frow[kk];                a.y = frow[kk + 1];
        v2f b; b.x = W1[kk * D_HID + l];      b.y = W1[(kk + 1) * D_HID + l];
        c = __builtin_amdgcn_wmma_f32_16x16x4_f32(false, a, false, b,
                                                  (short)0, c, false, false);
    }
#pragma unroll
    for (int v = 0; v < 8; ++v) {
        int r = row0 + v + 8 * half;         // C/D layout: M = v + 8*half, N = l
        if (r < n_nodes) H1[(size_t)r * D_HID + l] = c[v];
    }
}

// ---------------------------------------------------------------- edge scatter
// agg[dst[e]*16 + d] += h[src[e]*16 + d]; one lane per (edge, dim) element.
// 16-dim tables are L2-resident; fp32 HW atomics.
__global__ void gcn_scatter(const float* __restrict__ h,
                            const int* __restrict__ src,
                            const int* __restrict__ dst,
                            float* __restrict__ agg, int n_elem) {
    int i = blockIdx.x * blockDim.x + threadIdx.x;
    if (i >= n_elem) return;
    int e = i >> 4;
    int d = i & 15;
    float v = h[(size_t)src[e] * D_HID + d];
    unsafeAtomicAdd(&agg[(size_t)dst[e] * D_HID + d], v);
}

// ---------------------------------------------------------------- bias + ReLU
__global__ void gcn_bias_relu(float* __restrict__ x, const float* __restrict__ b,
                              int n_elem) {
    int i = blockIdx.x * blockDim.x + threadIdx.x;
    if (i < n_elem) {
        float v = x[i] + b[i & 15];
        x[i] = v > 0.0f ? v : 0.0f;
    }
}

// ---------------------------------------------------------------- layer-2 GEMM
// out[n x 32] = A[n x 16] @ W2[16 x 32] + b2; two 16x16 N-tiles per wave.
__global__ __launch_bounds__(256) void gcn_gemm_l2(const float* __restrict__ A,
                                                   const float* __restrict__ W2,
                                                   const float* __restrict__ b2,
                                                   float* __restrict__ out,
                                                   int n_nodes) {
    int wave = (blockIdx.x * blockDim.x + threadIdx.x) >> 5;
    int lane = threadIdx.x & 31;
    int half = lane >> 4;
    int l    = lane & 15;
    int row0 = wave * 16;
    if (row0 >= n_nodes) return;

    int arow = row0 + l;
    if (arow >= n_nodes) arow = n_nodes - 1;
    const float* ap = A + (size_t)arow * D_HID;

    v8f c0 = {}, c1 = {};
#pragma unroll
    for (int k = 0; k < D_HID; k += 4) {
        int kk = k + 2 * half;
        v2f a;  a.x  = ap[kk];                     a.y  = ap[kk + 1];
        v2f b0; b0.x = W2[kk * D_OUT + l];         b0.y = W2[(kk + 1) * D_OUT + l];
        v2f b1; b1.x = W2[kk * D_OUT + 16 + l];    b1.y = W2[(kk + 1) * D_OUT + 16 + l];
        c0 = __builtin_amdgcn_wmma_f32_16x16x4_f32(false, a, false, b0,
                                                   (short)0, c0, false, false);
        c1 = __builtin_amdgcn_wmma_f32_16x16x4_f32(false, a, false, b1,
                                                   (short)0, c1, false, false);
    }
    float bl0 = b2[l], bl1 = b2[16 + l];
#pragma unroll
    for (int v = 0; v < 8; ++v) {
        int r = row0 + v + 8 * half;
        if (r < n_nodes) {
            out[(size_t)r * D_OUT + l]      = c0[v] + bl0;
            out[(size_t)r * D_OUT + 16 + l] = c1[v] + bl1;
        }
    }
}

// ---------------------------------------------------------------- launch
extern "C" void kernel_launch(void* const* d_in, const int* in_sizes, int n_in,
                              void* d_out, int out_size, void* d_ws, size_t ws_size,
                              hipStream_t stream) {
    const float* features = (const float*)d_in[0];
    const int*   src      = (const int*)d_in[1];
    const int*   dst      = (const int*)d_in[2];
    const float* W1       = (const float*)d_in[3];
    const float* b1       = (const float*)d_in[4];
    const float* W2       = (const float*)d_in[5];
    const float* b2       = (const float*)d_in[6];
    float*       out      = (float*)d_out;

    int n_nodes = in_sizes[0] / D_IN;
    int n_edges = in_sizes[1];
    size_t nh   = (size_t)n_nodes * D_HID;

    float* H1   = (float*)d_ws;   // n_nodes x 16
    float* agg1 = H1 + nh;        // n_nodes x 16 (becomes x after bias+ReLU)
    float* agg2 = agg1 + nh;      // n_nodes x 16

    // zero both accumulators (contiguous block) every call — deterministic
    int nz = (int)(2 * nh);
    gcn_zero<<<(nz + 255) / 256, 256, 0, stream>>>(agg1, nz);

    int ntiles  = (n_nodes + 15) / 16;
    int nblocks = (ntiles + 7) / 8;          // 8 waves (256 threads) per block

    // layer 1: transform BEFORE aggregation (8x less edge traffic)
    gcn_gemm_l1<<<nblocks, 256, 0, stream>>>(features, W1, H1, n_nodes);

    int nelem = n_edges * D_HID;             // 25.6M
    gcn_scatter<<<(nelem + 255) / 256, 256, 0, stream>>>(H1, src, dst, agg1, nelem);

    gcn_bias_relu<<<((int)nh + 255) / 256, 256, 0, stream>>>(agg1, b1, (int)nh);

    // layer 2: aggregate 16-dim activations, then project to 32
    gcn_scatter<<<(nelem + 255) / 256, 256, 0, stream>>>(agg1, src, dst, agg2, nelem);

    gcn_gemm_l2<<<nblocks, 256, 0, stream>>>(agg2, W2, b2, out, n_nodes);
}